// PosMLP_14061722927240
// MI455X (gfx1250) — compile-verified
//
#include <hip/hip_runtime.h>

typedef __attribute__((ext_vector_type(2))) float v2f;
typedef __attribute__((ext_vector_type(8))) float v8f;

#define HD 16  // HIDDEN_DIM in the reference (n_w = 4*HD+1 = 65)

// ---------------------------------------------------------------------------
// Kernel 1: weights[M x N] = Q[M x C] * Wg[N x C]^T + b[N]
// Full-f32 GEMM via V_WMMA_F32_16X16X4_F32 (CDNA5-only op, no precision loss).
// One wave per 16x16 output tile; K stepped by 4 (C=256 -> 64 WMMA issues).
//
// Fragment layouts (ISA 7.12.2, wave32):
//   A 16x4:  lanes 0-15 -> M=lane,   v0=K0 v1=K1 ; lanes 16-31 -> v0=K2 v1=K3
//   B 4x16:  lanes 0-15 -> N=lane,   v0=K0 v1=K1 ; lanes 16-31 -> v0=K2 v1=K3
//   C/D:     VGPR i -> M = i + 8*(lane>=16), N = lane&15
//
// OOB handling: D row m depends only on A row m; D col n only on B col n.
// So OOB lanes just CLAMP their row pointer to valid memory (no zero-fill,
// no per-load predication) and the corrupted D elements are simply not
// stored. This keeps the inner loop branch-free: 2x global_load_b64 + wmma.
// ---------------------------------------------------------------------------
__global__ void posmlp_gen_wmma(const float* __restrict__ Q,
                                const float* __restrict__ Wg,
                                const float* __restrict__ bg,
                                float* __restrict__ Wout,
                                int M, int N, int C, int tilesN, int nTiles) {
  const int wavesPerBlk = blockDim.x >> 5;
  const int tile = blockIdx.x * wavesPerBlk + ((int)threadIdx.x >> 5);
  if (tile >= nTiles) return;  // wave-uniform exit: EXEC stays all-1s for WMMA

  const int tm = tile / tilesN;
  const int tn = tile - tm * tilesN;
  const int m0 = tm * 16, n0 = tn * 16;

  const int lane = threadIdx.x & 31;
  const int half = lane >> 4;   // 0: holds K={0,1}; 1: holds K={2,3}
  const int l    = lane & 15;

  const int mA = m0 + l;        // A-fragment row held by this lane
  const int nB = n0 + l;        // B-fragment col held by this lane
  const int mC = (mA < M) ? mA : (M - 1);   // clamp: load valid memory only
  const int nC = (nB < N) ? nB : (N - 1);

  // Each lane reads 2 K-contiguous floats per step -> one b64 load each.
  const float* qrow = Q  + (size_t)mC * C + 2 * half;
  const float* wrow = Wg + (size_t)nC * C + 2 * half;

  v8f acc = {};
  #pragma unroll 4
  for (int k = 0; k < C; k += 4) {   // C is a multiple of 4 (C = 256)
    const v2f a = *(const v2f*)(qrow + k);
    const v2f b = *(const v2f*)(wrow + k);
    // 8 args: (neg_a, A, neg_b, B, c_mod, C, reuse_a, reuse_b)
    acc = __builtin_amdgcn_wmma_f32_16x16x4_f32(
        false, a, false, b, (short)0, acc, false, false);
  }

  const float bias = bg[nC];
  const bool  nOk  = (nB < N);
  #pragma unroll
  for (int i = 0; i < 8; ++i) {
    const int m = m0 + i + 8 * half;
    if (m < M && nOk) Wout[(size_t)m * N + nB] = acc[i] + bias;
  }
}

// ---------------------------------------------------------------------------
// Kernel 2: per-query spatial MLP evaluation.
// One block per (b,q). Generated weights staged LDS -> registers; inner K=16
// loop fully unrolled (pure FMA/MAX VALU, no LDS in the hot loop).
// out[bq, h, w] = sum_k relu(rel_y*w1y[k] + rel_x*w1x[k] + b1[k]) * w2[k] + b2
// ---------------------------------------------------------------------------
__global__ void posmlp_spatial(const float* __restrict__ Wq,   // [BQ x N]
                               const float* __restrict__ pos,  // [BQ x 4]
                               const int* __restrict__ Hp,
                               const int* __restrict__ Wp,
                               float* __restrict__ out,
                               int N, int HW) {
  __shared__ float sw[96];  // N = 4*HD+1 = 65 <= 96
  const int bq = blockIdx.x;
  const int t  = threadIdx.x;

  if (t < N) sw[t] = Wq[(size_t)bq * N + t];
  __syncthreads();

  const int H_ = *Hp;
  const int W_ = *Wp;

  float w1y[HD], w1x[HD], b1[HD], w2[HD];
  #pragma unroll
  for (int k = 0; k < HD; ++k) {
    w1y[k] = sw[2 * k];          // weights[..., :2*hd] reshaped (hd,2) -> [...,0]
    w1x[k] = sw[2 * k + 1];      // -> [...,1]
    b1[k]  = sw[2 * HD + k];
    w2[k]  = sw[3 * HD + k];
  }
  const float b2 = sw[4 * HD];

  const float cx = pos[bq * 4 + 0];
  const float cy = pos[bq * 4 + 1];
  const float invbw = 1.0f / pos[bq * 4 + 2];
  const float invbh = 1.0f / pos[bq * 4 + 3];
  const float invH = 1.0f / (float)H_;
  const float invW = 1.0f / (float)W_;

  float* obase = out + (size_t)bq * HW;
  for (int p = t; p < HW; p += blockDim.x) {
    const int h = p / W_;
    const int w = p - h * W_;
    const float rel_y = (((float)h + 0.5f) * invH - cy) * invbh;
    const float rel_x = (((float)w + 0.5f) * invW - cx) * invbw;
    float acc = b2;
    #pragma unroll
    for (int k = 0; k < HD; ++k) {
      const float hid = fmaf(rel_y, w1y[k], fmaf(rel_x, w1x[k], b1[k]));
      acc = fmaf(fmaxf(hid, 0.0f), w2[k], acc);
    }
    obase[p] = acc;
  }
}

// ---------------------------------------------------------------------------
// Launch: inputs in setup_inputs() order:
//   d_in[0] queries (B,Q,C) f32 | d_in[1] pos (B,Q,4) f32
//   d_in[2] w_gen (n_w,C) f32   | d_in[3] b_gen (n_w,) f32
//   d_in[4] H (int scalar)      | d_in[5] W (int scalar)
// d_out: (B,Q,H,W) f32. d_ws: intermediate weights (BQ*N floats = 468 KB).
// ---------------------------------------------------------------------------
extern "C" void kernel_launch(void* const* d_in, const int* in_sizes, int n_in,
                              void* d_out, int out_size, void* d_ws, size_t ws_size,
                              hipStream_t stream) {
  const float* queries = (const float*)d_in[0];
  const float* pos     = (const float*)d_in[1];
  const float* w_gen   = (const float*)d_in[2];
  const float* b_gen   = (const float*)d_in[3];
  const int*   Hp      = (const int*)d_in[4];
  const int*   Wp      = (const int*)d_in[5];

  const int N  = in_sizes[3];            // n_w = 65
  const int C  = in_sizes[2] / N;        // 256
  const int BQ = in_sizes[0] / C;        // B*Q = 1800
  const int HW = out_size / BQ;          // H*W = 2304

  float* Wout = (float*)d_ws;            // [BQ x N] intermediate

  const int tilesM = (BQ + 15) / 16;     // 113
  const int tilesN = (N + 15) / 16;      // 5
  const int nTiles = tilesM * tilesN;    // 565
  const int wavesPerBlk = 4;
  const int blocks1 = (nTiles + wavesPerBlk - 1) / wavesPerBlk;

  posmlp_gen_wmma<<<blocks1, wavesPerBlk * 32, 0, stream>>>(
      queries, w_gen, b_gen, Wout, BQ, N, C, tilesN, nTiles);

  posmlp_spatial<<<BQ, 256, 0, stream>>>(
      Wout, pos, Hp, Wp, (float*)d_out, N, HW);
}